// PointNetFeaturePropagation_60507499266141
// MI455X (gfx1250) — compile-verified
//
#include <hip/hip_runtime.h>

// ---------------------------------------------------------------------------
// PointNet Feature Propagation for MI455X (gfx1250, wave32, WMMA).
// Pipeline: topk(3-NN) -> build X (concat+interp) -> WMMA fp32 GEMM1 ->
//           BN stats/normalize+ReLU -> WMMA fp32 GEMM2 -> BN+ReLU on d_out.
// GEMM tiles staged with GLOBAL_LOAD_ASYNC_TO_LDS_B128 (ASYNCcnt) and
// double-buffered LDS so DMA overlaps v_wmma_f32_16x16x4_f32 issue.
// ---------------------------------------------------------------------------

typedef __attribute__((ext_vector_type(2))) float v2f;
typedef __attribute__((ext_vector_type(4))) float v4f;
typedef __attribute__((ext_vector_type(8))) float v8f;

#define EPS_INTERP 1e-8f
#define EPS_BN 1e-5f

constexpr int B_   = 8;
constexpr int N_   = 8192;
constexpr int S_   = 2048;
constexpr int D1_  = 128;
constexpr int D2_  = 256;
constexpr int R_   = B_ * N_;     // 65536 rows
constexpr int CIN_ = D1_ + D2_;   // 384
constexpr int C1_  = 256;
constexpr int C2_  = 128;

// Async DMA of 16 bytes global -> LDS, tracked with ASYNCcnt.
// VDST = 32-bit LDS byte address (low 32 bits of the generic pointer are the
// LDS offset per the flat-addressing aperture rules), VADDR = 64-bit global.
__device__ __forceinline__ void async_copy_b128(const float* g, const float* lds) {
  unsigned loff = (unsigned)(unsigned long long)lds;
  asm volatile("global_load_async_to_lds_b128 %0, %1, off"
               :: "v"(loff), "v"(g)
               : "memory");
}

// ---------------------------------------------------------------------------
// Kernel 1: 3-nearest-neighbor search. One thread per query point, xyz2
// staged through LDS in 512-point chunks (6 KB).
// ---------------------------------------------------------------------------
__global__ __launch_bounds__(256) void topk_kernel(
    const float* __restrict__ xyz1, const float* __restrict__ xyz2,
    int* __restrict__ idx, float* __restrict__ wts) {
  __shared__ float sx[512 * 3];
  const int b = blockIdx.y;
  const int n = blockIdx.x * 256 + threadIdx.x;

  const float x = xyz1[(b * N_ + n) * 3 + 0];
  const float y = xyz1[(b * N_ + n) * 3 + 1];
  const float z = xyz1[(b * N_ + n) * 3 + 2];

  float d0 = 1e30f, d1 = 1e30f, d2 = 1e30f;
  int i0 = 0, i1 = 0, i2 = 0;

  for (int s0 = 0; s0 < S_; s0 += 512) {
    __syncthreads();
#pragma unroll
    for (int q = 0; q < 6; ++q) {
      int t = threadIdx.x + q * 256;
      sx[t] = xyz2[b * S_ * 3 + s0 * 3 + t];
    }
    __syncthreads();
    for (int s = 0; s < 512; ++s) {
      float dx = x - sx[s * 3 + 0];
      float dy = y - sx[s * 3 + 1];
      float dz = z - sx[s * 3 + 2];
      float d = dx * dx + dy * dy + dz * dz;
      int gs = s0 + s;
      if (d < d2) {
        if (d < d1) {
          d2 = d1; i2 = i1;
          if (d < d0) { d1 = d0; i1 = i0; d0 = d; i0 = gs; }
          else        { d1 = d;  i1 = gs; }
        } else { d2 = d; i2 = gs; }
      }
    }
  }

  float r0 = 1.f / (d0 + EPS_INTERP);
  float r1 = 1.f / (d1 + EPS_INTERP);
  float r2 = 1.f / (d2 + EPS_INTERP);
  float rs = 1.f / (r0 + r1 + r2);
  int base = (b * N_ + n) * 3;
  idx[base + 0] = i0; idx[base + 1] = i1; idx[base + 2] = i2;
  wts[base + 0] = r0 * rs; wts[base + 1] = r1 * rs; wts[base + 2] = r2 * rs;
}

// ---------------------------------------------------------------------------
// Kernel 2: build X[r, 0:128] = points1, X[r, 128:384] = 3-NN interpolation.
// One thread per 4 consecutive channels (rows are multiples of 4 channels),
// b128 traffic on points1 / points2 / X.
// ---------------------------------------------------------------------------
__global__ __launch_bounds__(256) void build_x_kernel(
    const float* __restrict__ p1, const float* __restrict__ p2,
    const int* __restrict__ idx, const float* __restrict__ wts,
    float* __restrict__ X) {
  int i4 = blockIdx.x * 256 + threadIdx.x;  // < R_*CIN_/4
  int i = i4 * 4;
  int row = i / CIN_;
  int c = i - row * CIN_;
  v4f v;
  if (c < D1_) {
    v = *(const v4f*)&p1[row * D1_ + c];
  } else {
    int cc = c - D1_;
    int b = row >> 13;                       // N_ = 8192
    int base = row * 3;
    const float* p2b = p2 + b * (S_ * D2_);
    float w0 = wts[base + 0], w1 = wts[base + 1], w2 = wts[base + 2];
    v4f g0 = *(const v4f*)&p2b[idx[base + 0] * D2_ + cc];
    v4f g1 = *(const v4f*)&p2b[idx[base + 1] * D2_ + cc];
    v4f g2 = *(const v4f*)&p2b[idx[base + 2] * D2_ + cc];
    v = w0 * g0 + w1 * g1 + w2 * g2;
  }
  *(v4f*)&X[i] = v;
}

// ---------------------------------------------------------------------------
// Kernel 3: fp32 WMMA GEMM:  Y[r][o] = sum_k A[r][k] * W[o][k] + bias[o].
// Block = 256 threads = 8 wave32s; block tile 64(M) x 128(N); each wave owns
// a 32x32 tile = 2x2 V_WMMA_F32_16X16X4_F32 accumulators.
// Tiles staged with async global->LDS b128 DMA, double-buffered: prefetch
// chunk k+1 into buf^1, s_wait_asynccnt 3 leaves only the prefetch in
// flight (async loads complete in order), barrier, then 4 WMMA k-steps (16 WMMAs).
// LDS stride 20 floats: conflict-free b64 fragment reads, 16B-aligned fills.
// Fragment layout per ISA 7.12.2 (fp32 16x4): lane half h, vgpr j -> K=2h+j.
// ---------------------------------------------------------------------------
template <int K, int NC>
__global__ __launch_bounds__(256) void gemm_wmma_kernel(
    const float* __restrict__ A, const float* __restrict__ W,
    const float* __restrict__ bias, float* __restrict__ Y) {
  __shared__ float As[2][64 * 20];
  __shared__ float Bs[2][128 * 20];

  const int tid  = threadIdx.x;
  const int wave = tid >> 5;
  const int lane = tid & 31;
  const int lm   = lane & 15;
  const int h    = lane >> 4;
  const int wm   = wave >> 2;   // 0..1  (M)
  const int wn   = wave & 3;    // 0..3  (N)

  const int rowBase = blockIdx.x * 64;
  const int colBase = blockIdx.y * 128;

  v8f acc[2][2] = {};

  // A-tile: one b128 per thread; B-tile: two b128 per thread.
  const int ai  = tid >> 2;             // A row 0..63
  const int ak  = (tid & 3) * 4;        // A col {0,4,8,12}
  const int bi0 = (tid * 2) >> 2;       // B row 0..127
  const int bk0 = ((tid * 2) & 3) * 4;
  const int bi1 = (tid * 2 + 1) >> 2;
  const int bk1 = ((tid * 2 + 1) & 3) * 4;

  const float* gA = &A[(rowBase + ai) * K + ak];
  const float* gB0 = &W[(colBase + bi0) * K + bk0];
  const float* gB1 = &W[(colBase + bi1) * K + bk1];

  // prefetch chunk 0 into buffer 0
  async_copy_b128(gA, &As[0][ai * 20 + ak]);
  async_copy_b128(gB0, &Bs[0][bi0 * 20 + bk0]);
  async_copy_b128(gB1, &Bs[0][bi1 * 20 + bk1]);

  int buf = 0;
  for (int k0 = 0; k0 < K; k0 += 16, buf ^= 1) {
    if (k0 + 16 < K) {
      // prefetch next chunk into the other buffer, then wait for current
      async_copy_b128(gA + k0 + 16, &As[buf ^ 1][ai * 20 + ak]);
      async_copy_b128(gB0 + k0 + 16, &Bs[buf ^ 1][bi0 * 20 + bk0]);
      async_copy_b128(gB1 + k0 + 16, &Bs[buf ^ 1][bi1 * 20 + bk1]);
      asm volatile("s_wait_asynccnt 0x3" ::: "memory");
    } else {
      asm volatile("s_wait_asynccnt 0x0" ::: "memory");
    }
    __syncthreads();

    const float* Asb = As[buf];
    const float* Bsb = Bs[buf];
#pragma unroll
    for (int kk = 0; kk < 16; kk += 4) {
      v2f a0 = *(const v2f*)&Asb[(wm * 32 + 0 + lm) * 20 + kk + 2 * h];
      v2f a1 = *(const v2f*)&Asb[(wm * 32 + 16 + lm) * 20 + kk + 2 * h];
      v2f b0 = *(const v2f*)&Bsb[(wn * 32 + 0 + lm) * 20 + kk + 2 * h];
      v2f b1 = *(const v2f*)&Bsb[(wn * 32 + 16 + lm) * 20 + kk + 2 * h];
      acc[0][0] = __builtin_amdgcn_wmma_f32_16x16x4_f32(
          false, a0, false, b0, (short)0, acc[0][0], false, false);
      acc[0][1] = __builtin_amdgcn_wmma_f32_16x16x4_f32(
          false, a0, false, b1, (short)0, acc[0][1], false, false);
      acc[1][0] = __builtin_amdgcn_wmma_f32_16x16x4_f32(
          false, a1, false, b0, (short)0, acc[1][0], false, false);
      acc[1][1] = __builtin_amdgcn_wmma_f32_16x16x4_f32(
          false, a1, false, b1, (short)0, acc[1][1], false, false);
    }
    __syncthreads();  // all waves done reading buf before it is refilled
  }

  // Epilogue: D layout — vgpr r holds rows M=r (lanes 0-15) and M=8+r (16-31).
#pragma unroll
  for (int tn = 0; tn < 2; ++tn) {
    int col = colBase + wn * 32 + tn * 16 + lm;
    float bv = bias[col];
#pragma unroll
    for (int tm = 0; tm < 2; ++tm) {
      int rbase = rowBase + wm * 32 + tm * 16 + 8 * h;
#pragma unroll
      for (int r = 0; r < 8; ++r) {
        Y[(rbase + r) * NC + col] = acc[tm][tn][r] + bv;
      }
    }
  }
}

// ---------------------------------------------------------------------------
// Kernel 4: per-channel sum / sum-of-squares (batch-norm batch statistics).
// blockDim.x == C; each block reduces 256 rows, then float global atomics.
// ---------------------------------------------------------------------------
template <int C>
__global__ void colsum_kernel(const float* __restrict__ Y,
                              float* __restrict__ sum, float* __restrict__ ss) {
  int c = threadIdx.x;
  int r0 = blockIdx.x * 256;
  float s = 0.f, q = 0.f;
  for (int r = 0; r < 256; ++r) {
    float v = Y[(r0 + r) * C + c];
    s += v;
    q += v * v;
  }
  atomicAdd(&sum[c], s);
  atomicAdd(&ss[c], q);
}

// ---------------------------------------------------------------------------
// Kernel 5: finalize BN: scale = g*rsqrt(var+eps), shift = be - mu*scale.
// ---------------------------------------------------------------------------
__global__ void bnfin_kernel(const float* __restrict__ sum,
                             const float* __restrict__ ss,
                             const float* __restrict__ g,
                             const float* __restrict__ be,
                             float* __restrict__ scale,
                             float* __restrict__ shift) {
  int c = threadIdx.x;
  float inv = 1.f / (float)R_;
  float mu = sum[c] * inv;
  float var = ss[c] * inv - mu * mu;
  float sc = g[c] * rsqrtf(var + EPS_BN);
  scale[c] = sc;
  shift[c] = be[c] - mu * sc;
}

// ---------------------------------------------------------------------------
// Kernel 6: in-place normalize + ReLU, 4 channels per thread (b128).
// ---------------------------------------------------------------------------
template <int C>
__global__ __launch_bounds__(256) void bnrelu_kernel(
    float* __restrict__ Y, const float* __restrict__ scale,
    const float* __restrict__ shift) {
  int i = (blockIdx.x * 256 + threadIdx.x) * 4;
  int c = i & (C - 1);
  v4f v = *(const v4f*)&Y[i];
  v4f sc = *(const v4f*)&scale[c];
  v4f sh = *(const v4f*)&shift[c];
  v = v * sc + sh;
#pragma unroll
  for (int q = 0; q < 4; ++q) v[q] = fmaxf(v[q], 0.f);
  *(v4f*)&Y[i] = v;
}

// ---------------------------------------------------------------------------
// Host side
// ---------------------------------------------------------------------------
extern "C" void kernel_launch(void* const* d_in, const int* in_sizes, int n_in,
                              void* d_out, int out_size, void* d_ws, size_t ws_size,
                              hipStream_t stream) {
  const float* xyz1 = (const float*)d_in[0];
  const float* xyz2 = (const float*)d_in[1];
  const float* p1   = (const float*)d_in[2];
  const float* p2   = (const float*)d_in[3];
  const float* w1   = (const float*)d_in[4];
  const float* b1   = (const float*)d_in[5];
  const float* g1   = (const float*)d_in[6];
  const float* be1  = (const float*)d_in[7];
  const float* w2   = (const float*)d_in[8];
  const float* b2   = (const float*)d_in[9];
  const float* g2   = (const float*)d_in[10];
  const float* be2  = (const float*)d_in[11];
  float* out = (float*)d_out;

  // Workspace layout (bytes, all 256B aligned):
  const size_t X_BYTES  = (size_t)R_ * CIN_ * sizeof(float);   // 100,663,296
  const size_t Y1_BYTES = (size_t)R_ * C1_ * sizeof(float);    //  67,108,864
  const size_t I_BYTES  = (size_t)R_ * 3 * sizeof(int);        //     786,432
  char* ws = (char*)d_ws;
  float* X   = (float*)(ws);
  float* Y1  = (float*)(ws + X_BYTES);
  int*   idx = (int*)  (ws + X_BYTES + Y1_BYTES);
  float* wts = (float*)(ws + X_BYTES + Y1_BYTES + I_BYTES);
  float* st  = (float*)(ws + X_BYTES + Y1_BYTES + 2 * I_BYTES);
  // stats block: [sum1 256][ss1 256][sum2 128][ss2 128][scale1 256][shift1 256][scale2 128][shift2 128]
  float* sum1 = st;
  float* ss1  = st + 256;
  float* sum2 = st + 512;
  float* ss2  = st + 640;
  float* scale1 = st + 768;
  float* shift1 = st + 1024;
  float* scale2 = st + 1280;
  float* shift2 = st + 1408;

  // Zero the accumulated stats every call (ws is NOT re-poisoned per replay).
  hipMemsetAsync(st, 0, 768 * sizeof(float), stream);

  // 1. three-NN
  topk_kernel<<<dim3(N_ / 256, B_), 256, 0, stream>>>(xyz1, xyz2, idx, wts);

  // 2. concat + interpolate -> X [R_, 384]
  build_x_kernel<<<(R_ * CIN_ / 4) / 256, 256, 0, stream>>>(p1, p2, idx, wts, X);

  // 3. GEMM1 (fp32 WMMA, async LDS staging): Y1 = X @ w1^T + b1
  gemm_wmma_kernel<CIN_, C1_><<<dim3(R_ / 64, C1_ / 128), 256, 0, stream>>>(X, w1, b1, Y1);

  // 4. BN1 stats + normalize + ReLU (in place on Y1)
  colsum_kernel<C1_><<<R_ / 256, C1_, 0, stream>>>(Y1, sum1, ss1);
  bnfin_kernel<<<1, C1_, 0, stream>>>(sum1, ss1, g1, be1, scale1, shift1);
  bnrelu_kernel<C1_><<<(R_ * C1_ / 4) / 256, 256, 0, stream>>>(Y1, scale1, shift1);

  // 5. GEMM2 (fp32 WMMA): out = Y1 @ w2^T + b2   (pre-BN, into d_out)
  gemm_wmma_kernel<C1_, C2_><<<dim3(R_ / 64, C2_ / 128), 256, 0, stream>>>(Y1, w2, b2, out);

  // 6. BN2 stats + normalize + ReLU (in place on d_out)
  colsum_kernel<C2_><<<R_ / 256, C2_, 0, stream>>>(out, sum2, ss2);
  bnfin_kernel<<<1, C2_, 0, stream>>>(sum2, ss2, g2, be2, scale2, shift2);
  bnrelu_kernel<C2_><<<(R_ * C2_ / 4) / 256, 256, 0, stream>>>(out, scale2, shift2);
}